// NCC_map_46308337385681
// MI455X (gfx1250) — compile-verified
//
#include <hip/hip_runtime.h>

typedef __attribute__((ext_vector_type(2))) float v2f;
typedef __attribute__((ext_vector_type(4))) float v4f;
typedef __attribute__((ext_vector_type(8))) float v8f;
typedef int v4i __attribute__((vector_size(16)));
typedef __attribute__((address_space(1))) v4i* gv4i_p;  // global (AS1)
typedef __attribute__((address_space(3))) v4i* lv4i_p;  // LDS (AS3)

#define N 192
#define TILE 16
#define PAD 4
#define PADDED 24          // TILE + 2*PAD
#define XSTR 32            // LDS row stride (cols 24..31 stay zero)
#define DCHUNK 48
#define SMAX (DCHUNK + 2 * PAD)   // 56 slices processed per task
#define NTH (N / TILE)     // 12
#define NCHUNK (N / DCHUNK)// 4
#define WPB 2              // waves per block

// per-wave LDS layout (in floats)
#define BUF_FLOATS (PADDED * XSTR)             // 768
#define BUFI_OFF 0                              // double-buffered: [2][768]
#define BUFJ_OFF (2 * BUF_FLOATS)               // 1536
#define TBUF_OFF (4 * BUF_FLOATS)               // 3072
#define DUMMY_OFF (TBUF_OFF + TILE * XSTR)      // 3584 (sink for masked async lanes)
#define RING_OFF (DUMMY_OFF + 16)               // 3600
#define RING_FLOATS (9 * 5 * 32 * 8)            // 11520
#define WAVE_FLOATS (RING_OFF + RING_FLOATS)    // 15120

#define HAS_ASYNC __has_builtin(__builtin_amdgcn_global_load_async_to_lds_b128)

__device__ __forceinline__ float band(int m, int j) {
  // ones-band: output position m sums padded positions [m, m+8]
  return (j >= m && j <= m + 8) ? 1.0f : 0.0f;
}

__device__ __forceinline__ float fieldv(int f, float a, float b) {
  if (f == 0) return a;
  if (f == 1) return b;
  if (f == 2) return a * a;
  if (f == 3) return b * b;
  return a * b;
}

#if HAS_ASYNC
__device__ __forceinline__ void async_cp16(const float* g, float* l) {
  __builtin_amdgcn_global_load_async_to_lds_b128((gv4i_p)g, (lv4i_p)l, 0, 0);
}
#endif

__global__ __launch_bounds__(WPB * 32) void ncc_map_kernel(
    const float* __restrict__ I, const float* __restrict__ J,
    float* __restrict__ O) {
  extern __shared__ float smem[];
  const int lane = threadIdx.x & 31;
  const int wave = threadIdx.x >> 5;
  const int g = lane >> 4;    // lane group (0/1) per WMMA fragment layout
  const int l16 = lane & 15;

  float* wls   = smem + wave * WAVE_FLOATS;
  float* bufI  = wls + BUFI_OFF;
  float* bufJ  = wls + BUFJ_OFF;
  float* Tbuf  = wls + TBUF_OFF;
  float* dummy = wls + DUMMY_OFF;
  float* ring  = wls + RING_OFF;

  const int task  = blockIdx.x * WPB + wave;          // 0..575
  const int chunk = task / (NTH * NTH);
  const int rem   = task % (NTH * NTH);
  const int h0 = (rem / NTH) * TILE;
  const int w0 = (rem % NTH) * TILE;
  const int d0 = chunk * DCHUNK;

#if HAS_ASYNC
  // One-time zero of both I/J double buffers + dummy: pad rows/cols and
  // cols 24..31 are never written by the async copies and must read as 0.
  for (int idx = lane; idx < TBUF_OFF; idx += 32) wls[idx] = 0.0f;
  for (int idx = lane; idx < 16; idx += 32) dummy[idx] = 0.0f;
  __syncthreads();

  // Prologue: issue async copy of slice 0 (dglob = d0-4). For chunk 0 the
  // slice is OOB and the zero-init already covers it.
  {
    const int dn = d0 - PAD;
    if ((unsigned)dn < (unsigned)N) {
      const int sbase = dn * (N * N);
#pragma unroll
      for (int it = 0; it < 5; ++it) {
        const int idx = it * 32 + lane;       // 0..159 over 24 rows x 6 b128
        const int i = idx / 6, j4 = idx - i * 6;
        const int gh = h0 - PAD + i;
        const int gw = w0 - PAD + 4 * j4;
        const bool ok = (idx < 144) && ((unsigned)gh < (unsigned)N) &&
                        ((unsigned)gw < (unsigned)N);
        const int gi = ok ? (sbase + gh * N + gw) : 0;
        float* lI = ok ? (bufI + i * XSTR + 4 * j4) : dummy;
        float* lJ = ok ? (bufJ + i * XSTR + 4 * j4) : dummy;
        async_cp16(I + gi, lI);
        async_cp16(J + gi, lJ);
      }
    }
  }
#endif

  for (int s = 0; s < SMAX; ++s) {
#if HAS_ASYNC
    // ---- prefetch slice s+1 into the other buffer (10 async b128, always) ----
    bool nextAsync = false;
    if (s + 1 < SMAX) {
      const int dn = d0 - PAD + s + 1;
      float* bIn = bufI + ((s + 1) & 1) * BUF_FLOATS;
      float* bJn = bufJ + ((s + 1) & 1) * BUF_FLOATS;
      if ((unsigned)dn < (unsigned)N) {
        const int sbase = dn * (N * N);
#pragma unroll
        for (int it = 0; it < 5; ++it) {
          const int idx = it * 32 + lane;
          const int i = idx / 6, j4 = idx - i * 6;
          const int gh = h0 - PAD + i;
          const int gw = w0 - PAD + 4 * j4;
          const bool ok = (idx < 144) && ((unsigned)gh < (unsigned)N) &&
                          ((unsigned)gw < (unsigned)N);
          const int gi = ok ? (sbase + gh * N + gw) : 0;
          float* lI = ok ? (bIn + i * XSTR + 4 * j4) : dummy;
          float* lJ = ok ? (bJn + i * XSTR + 4 * j4) : dummy;
          async_cp16(I + gi, lI);
          async_cp16(J + gi, lJ);
        }
        nextAsync = true;
      } else {
        // OOB slice: zero the in-range region (6 v4f per lane, exact)
        v4f z = {0.0f, 0.0f, 0.0f, 0.0f};
#pragma unroll
        for (int it = 0; it < 6; ++it) {
          const int q = (it * 32 + lane) * 4;
          *(v4f*)(bIn + q) = z;
          *(v4f*)(bJn + q) = z;
        }
      }
    }
    // Wait for CURRENT slice's async copies: async loads complete in order,
    // so allowing the 10 just-issued next-slice copies to stay outstanding
    // guarantees the current buffer is fully written.
    if (nextAsync) asm volatile("s_wait_asynccnt 0xa" ::: "memory");
    else           asm volatile("s_wait_asynccnt 0x0" ::: "memory");
    const float* bI = bufI + (s & 1) * BUF_FLOATS;
    const float* bJ = bufJ + (s & 1) * BUF_FLOATS;
#else
    // ---- synchronous fallback: load padded 24x24 I/J tile (zero-padded) ----
    __syncthreads();  // WAR vs previous slice's reads
    const int dglob_s = d0 - PAD + s;
    const bool inD = (unsigned)dglob_s < (unsigned)N;
    const int sbase = dglob_s * (N * N);
    float* bIw = bufI + (s & 1) * BUF_FLOATS;
    float* bJw = bufJ + (s & 1) * BUF_FLOATS;
#pragma unroll
    for (int it = 0; it < (PADDED * XSTR) / 32; ++it) {
      const int idx = it * 32 + lane;
      const int i = idx >> 5;
      const int j = idx & 31;
      const int gh = h0 - PAD + i;
      const int gw = w0 - PAD + j;
      const bool ok = inD && (j < PADDED) && ((unsigned)gh < (unsigned)N) &&
                      ((unsigned)gw < (unsigned)N);
      const int gi = sbase + gh * N + gw;
      bIw[idx] = ok ? I[gi] : 0.0f;
      bJw[idx] = ok ? J[gi] : 0.0f;
    }
    const float* bI = bIw;
    const float* bJ = bJw;
#endif

    const int slot = s % 9;

    // ---- per field: 2D (H,W) box sum via V_WMMA_F32_16X16X4_F32 ----
#pragma unroll
    for (int f = 0; f < 5; ++f) {
      __syncthreads();  // bufI/J visible; Tbuf WAR vs previous field
      // Step 1: T(16x24) = S(16x24) * X(24x24)  (H-axis 9-tap sum)
#pragma unroll
      for (int cb = 0; cb < 2; ++cb) {
        const int c0 = cb * 16;
        v8f acc = {};
#pragma unroll
        for (int k = 0; k < 6; ++k) {
          const int j0 = 4 * k + 2 * g;  // K index held by this lane's vgpr pair
          v2f a, b;
          a.x = band(l16, j0);
          a.y = band(l16, j0 + 1);
          const int col = c0 + l16;
          const float i0v = bI[j0 * XSTR + col];
          const float j0v = bJ[j0 * XSTR + col];
          const float i1v = bI[(j0 + 1) * XSTR + col];
          const float j1v = bJ[(j0 + 1) * XSTR + col];
          b.x = fieldv(f, i0v, j0v);
          b.y = fieldv(f, i1v, j1v);
          acc = __builtin_amdgcn_wmma_f32_16x16x4_f32(false, a, false, b,
                                                      (short)0, acc, false,
                                                      false);
        }
#pragma unroll
        for (int v = 0; v < 8; ++v)  // D-frag: row = v + 8*g, col = l16
          Tbuf[(v + 8 * g) * XSTR + c0 + l16] = acc[v];
      }
      __syncthreads();  // Tbuf visible
      // Step 2: Out(16x16) = T(16x24) * SwT(24x16)  (W-axis 9-tap sum)
      v8f o = {};
#pragma unroll
      for (int k = 0; k < 6; ++k) {
        const int j0 = 4 * k + 2 * g;
        v2f a = *(const v2f*)(Tbuf + l16 * XSTR + j0);  // 8B-aligned b64 load
        v2f b;
        b.x = band(l16, j0);      // SwT[j][w] = band(w, j)
        b.y = band(l16, j0 + 1);
        o = __builtin_amdgcn_wmma_f32_16x16x4_f32(false, a, false, b, (short)0,
                                                  o, false, false);
      }
      // lane-private ring store (same lane re-reads it: no barrier needed)
      v4f lo = {o[0], o[1], o[2], o[3]};
      v4f hi = {o[4], o[5], o[6], o[7]};
      v4f* rp = (v4f*)(ring + ((slot * 5 + f) * 32 + lane) * 8);
      rp[0] = lo;
      rp[1] = hi;
    }

    // ---- D-axis 9-tap sum + fused NCC combine, one output slice ----
    if (s >= 2 * PAD) {
      const int dout = d0 + s - 2 * PAD;
      v4f aLo[5], aHi[5];
#pragma unroll
      for (int f = 0; f < 5; ++f) {
        v4f z = {0.0f, 0.0f, 0.0f, 0.0f};
        aLo[f] = z;
        aHi[f] = z;
      }
#pragma unroll
      for (int sl = 0; sl < 9; ++sl) {
#pragma unroll
        for (int f = 0; f < 5; ++f) {
          const v4f* rp = (const v4f*)(ring + ((sl * 5 + f) * 32 + lane) * 8);
          aLo[f] += rp[0];
          aHi[f] += rp[1];
        }
      }
      const float inv = 1.0f / 729.0f;
#pragma unroll
      for (int v = 0; v < 8; ++v) {
        float Is, Js, I2, J2, IJ;
        if (v < 4) {
          Is = aLo[0][v]; Js = aLo[1][v]; I2 = aLo[2][v];
          J2 = aLo[3][v]; IJ = aLo[4][v];
        } else {
          const int u = v - 4;
          Is = aHi[0][u]; Js = aHi[1][u]; I2 = aHi[2][u];
          J2 = aHi[3][u]; IJ = aHi[4][u];
        }
        const float cross = IJ - Is * Js * inv;
        const float Ivar  = I2 - Is * Is * inv;
        const float Jvar  = J2 - Js * Js * inv;
        const float cc = cross * cross / (Ivar * Jvar + 1e-5f);
        O[(dout * N + (h0 + v + 8 * g)) * N + (w0 + l16)] = cc;
      }
    }
  }
}

extern "C" void kernel_launch(void* const* d_in, const int* in_sizes, int n_in,
                              void* d_out, int out_size, void* d_ws,
                              size_t ws_size, hipStream_t stream) {
  (void)in_sizes; (void)n_in; (void)d_ws; (void)ws_size; (void)out_size;
  const float* I = (const float*)d_in[0];
  const float* J = (const float*)d_in[1];
  float* O = (float*)d_out;

  const int tasks = NCHUNK * NTH * NTH;        // 576
  const int blocks = tasks / WPB;              // 288
  const size_t shmem = (size_t)WPB * WAVE_FLOATS * sizeof(float);  // ~118 KB
  ncc_map_kernel<<<blocks, WPB * 32, shmem, stream>>>(I, J, O);
}